// PolicyNetworkID_40527311405352
// MI455X (gfx1250) — compile-verified
//
#include <hip/hip_runtime.h>

#define DIMS   4096
#define HID    4096
#define NQ     512
#define M_IN   64
#define M_TOT  1088
#define NK     513
#define KLDA   528
#define SIGMA_C 0.1f
#define BIG_C   1.0e9f
#define QP_ITERS 25

typedef float v2f __attribute__((ext_vector_type(2)));
typedef float v8f __attribute__((ext_vector_type(8)));
typedef unsigned int u32x4 __attribute__((ext_vector_type(4)));
typedef int i32x4 __attribute__((ext_vector_type(4)));
typedef int i32x8 __attribute__((ext_vector_type(8)));

// ---------------- MLP GEMV (M=1, memory-bound, coalesced over columns) ----------
// act: 0 = none, 1 = relu, 2 = negate (produces q = -raw directly)
__global__ void k_gemv(const float* __restrict__ vin, const float* __restrict__ W,
                       const float* __restrict__ bias, float* __restrict__ vout,
                       int K, int N, int act) {
  int j = blockIdx.x * blockDim.x + threadIdx.x;
  if (j >= N) return;
  float acc = bias[j];
  const float* wp = W + j;
  for (int i = 0; i < K; i += 4) {
    acc = fmaf(vin[i],     wp[(size_t)i * N],       acc);
    acc = fmaf(vin[i + 1], wp[(size_t)(i + 1) * N], acc);
    acc = fmaf(vin[i + 2], wp[(size_t)(i + 2) * N], acc);
    acc = fmaf(vin[i + 3], wp[(size_t)(i + 3) * N], acc);
  }
  if (act == 1) acc = fmaxf(acc, 0.0f);
  else if (act == 2) acc = -acc;
  vout[j] = acc;
}

// ---------------- Build G = [A; I; -I], h = [b; ub; -lb]; init x,y,s,z ----------
__global__ void k_build(const float* __restrict__ A, const float* __restrict__ b,
                        const float* __restrict__ ub, const float* __restrict__ lb,
                        float* __restrict__ G, float* __restrict__ h,
                        float* __restrict__ x, float* __restrict__ y,
                        float* __restrict__ s, float* __restrict__ z) {
  int idx = blockIdx.x * blockDim.x + threadIdx.x;
  if (idx < M_TOT * NQ) {
    int r = idx / NQ, c = idx % NQ;
    float v;
    if (r < M_IN)            v = A[idx];
    else if (r < M_IN + NQ)  v = (r - M_IN == c) ? 1.0f : 0.0f;
    else                     v = (r - M_IN - NQ == c) ? -1.0f : 0.0f;
    G[idx] = v;
  }
  if (idx < M_TOT) {
    h[idx] = (idx < M_IN) ? b[idx]
           : (idx < M_IN + NQ ? ub[idx - M_IN] : -lb[idx - M_IN - NQ]);
    s[idx] = 1.0f;
    z[idx] = 1.0f;
  }
  if (idx < NQ) x[idx] = 0.0f;
  if (idx == 0) y[0] = 0.0f;
}

// ---------------- r_in = G x + s - h ; r_eq = Aeq x - beq -----------------------
__global__ void k_rin(const float* __restrict__ G, const float* __restrict__ x,
                      const float* __restrict__ s, const float* __restrict__ h,
                      const float* __restrict__ Aeq, const float* __restrict__ beq,
                      float* __restrict__ r_in, float* __restrict__ r_eq) {
  int r = blockIdx.x * blockDim.x + threadIdx.x;
  if (r < M_TOT) {
    float acc = 0.0f;
    for (int c = 0; c < NQ; ++c) acc = fmaf(G[(size_t)r * NQ + c], x[c], acc);
    r_in[r] = acc + s[r] - h[r];
  }
  if (r == 0) {
    float acc = 0.0f;
    for (int c = 0; c < NQ; ++c) acc = fmaf(Aeq[c], x[c], acc);
    r_eq[0] = acc - beq[0];
  }
}

// ---------------- mu = s.z/m ; w = z/s ; rv = SIGMA*mu/s - z + w*r_in -----------
__global__ void k_muwrv(const float* __restrict__ s, const float* __restrict__ z,
                        const float* __restrict__ r_in,
                        float* __restrict__ w, float* __restrict__ rv) {
  __shared__ float red[1024];
  int tid = threadIdx.x, nt = blockDim.x;
  float p = 0.0f;
  for (int i = tid; i < M_TOT; i += nt) p += s[i] * z[i];
  red[tid] = p;
  __syncthreads();
  for (int off = nt >> 1; off > 0; off >>= 1) {
    if (tid < off) red[tid] += red[tid + off];
    __syncthreads();
  }
  float mu = red[0] / (float)M_TOT;
  for (int i = tid; i < M_TOT; i += nt) {
    float wi = z[i] / s[i];
    w[i] = wi;
    rv[i] = SIGMA_C * mu / s[i] - z[i] + wi * r_in[i];
  }
}

// ---------------- rhs = [-(x+q+Aeq^T y + G^T(z+rv)); -r_eq] ; KKT border --------
__global__ void k_rhs_border(const float* __restrict__ G, const float* __restrict__ z,
                             const float* __restrict__ rv, const float* __restrict__ x,
                             const float* __restrict__ q, const float* __restrict__ Aeq,
                             const float* __restrict__ y, const float* __restrict__ r_eq,
                             float* __restrict__ rhs, float* __restrict__ KKT) {
  int p = threadIdx.x;
  if (p < NQ) {
    float acc = 0.0f;
    for (int k = 0; k < M_TOT; ++k)
      acc = fmaf(G[(size_t)k * NQ + p], z[k] + rv[k], acc);
    rhs[p] = -(x[p] + q[p] + Aeq[p] * y[0] + acc);
    KKT[(size_t)p * KLDA + NQ] = Aeq[p];   // Aeq^T column
    KKT[(size_t)NQ * KLDA + p] = Aeq[p];   // Aeq row
  } else if (p == NQ) {
    rhs[NQ] = -r_eq[0];
    KKT[(size_t)NQ * KLDA + NQ] = -1e-8f;  // reg
  }
}

// ---------------- Gw = diag(w) * G ---------------------------------------------
__global__ void k_scale_g(const float* __restrict__ G, const float* __restrict__ w,
                          float* __restrict__ Gw) {
  int idx = blockIdx.x * blockDim.x + threadIdx.x;
  if (idx < M_TOT * NQ) Gw[idx] = G[idx] * w[idx / NQ];
}

// ---------------- TDM: DMA a 16(k) x 16(col) f32 tile into LDS -----------------
// Builds the D# per CDNA5 ISA ch.8: group0 = {count=1, lds_addr, global_addr,
// type=2}; group1 = {data_size=4B, tensor_dim0/1 = 16, tile_dim0/1 = 16,
// tensor_dim0_stride = row_stride}; 2-D tile so remaining groups are zero.
// 6-arg toolchain variant: (u32x4, i32x8, i32x4, i32x4, i32x8, i32 cpol).
__device__ __forceinline__ void tdm_load_tile16x16(const float* gsrc,
                                                   unsigned lds_byte_addr,
                                                   int row_stride_elems) {
  unsigned long long ga = (unsigned long long)(uintptr_t)gsrc;
  u32x4 g0;
  g0[0] = 1u;                                   // count=1 (valid user descriptor)
  g0[1] = lds_byte_addr;                        // D#.lds_addr
  g0[2] = (unsigned)(ga & 0xffffffffu);         // global_addr[31:0]
  g0[3] = (unsigned)((ga >> 32) & 0x01ffffffu)  // global_addr[56:32]
        | (2u << 30);                           // type = 2 ("image")
  i32x8 g1;
  g1[0] = (int)(2u << 16);                      // data_size = 2 -> 4 bytes
  g1[1] = (int)(16u << 16);                     // tensor_dim0 = 16
  g1[2] = (int)(16u << 16);                     // tensor_dim1[15:0] = 16
  g1[3] = (int)(16u << 16);                     // tile_dim0 = 16
  g1[4] = 16;                                   // tile_dim1 = 16
  g1[5] = row_stride_elems;                     // tensor_dim0_stride[31:0]
  g1[6] = 0;
  g1[7] = 0;
  i32x4 z4 = {0, 0, 0, 0};                      // groups 2/3 unused (2-D tile)
  i32x8 z8 = {0, 0, 0, 0, 0, 0, 0, 0};
  __builtin_amdgcn_tensor_load_to_lds(g0, g1, z4, z4, z8, 0);
}

// ---------------- GtWG = Gw^T @ G via V_WMMA_F32_16X16X4_F32 -------------------
// One wave per 16x16 output tile; per-wave double-buffered LDS tiles fed by the
// Tensor Data Mover so the next K-chunk's DMA overlaps the current chunk's four
// WMMAs. A-layout (16x4 f32): lanes 0-15 hold K=0,1; lanes 16-31 hold K=2,3.
// C/D f32 16x16: VGPR r = rows r / r+8. Diagonal +1 (the In term) fused in.
__global__ void __launch_bounds__(256) k_gtwg_wmma(const float* __restrict__ Gw,
                                                   const float* __restrict__ G,
                                                   float* __restrict__ KKT) {
  __shared__ float lds[8][2][2][16 * 16];   // [wave][buf][A|B][256] = 32 KB
  int wave = threadIdx.x >> 5;
  int lane = threadIdx.x & 31;
  int tile = blockIdx.x * 8 + wave;
  const int TPD = NQ / 16;                  // 32 tiles per dimension
  int tp = (tile / TPD) * 16;               // output row block (p)
  int tq = (tile % TPD) * 16;               // output col block (q)
  int hh  = lane >> 4;                      // half-wave -> K pair (0,1) vs (2,3)
  int l16 = lane & 15;

  float* Ab[2] = { lds[wave][0][0], lds[wave][1][0] };
  float* Bb[2] = { lds[wave][0][1], lds[wave][1][1] };
  unsigned aoff[2] = { (unsigned)(uintptr_t)Ab[0], (unsigned)(uintptr_t)Ab[1] };
  unsigned boff[2] = { (unsigned)(uintptr_t)Bb[0], (unsigned)(uintptr_t)Bb[1] };

  // Preload chunk 0
  tdm_load_tile16x16(Gw + tp, aoff[0], NQ);
  tdm_load_tile16x16(G  + tq, boff[0], NQ);

  v8f acc = {};
  int buf = 0;
  for (int k0 = 0; k0 < M_TOT; k0 += 16) {
    __builtin_amdgcn_s_wait_tensorcnt(0);   // chunk k0 resident in LDS
    int kn = k0 + 16;
    if (kn < M_TOT) {                       // overlap next chunk's DMA w/ compute
      tdm_load_tile16x16(Gw + (size_t)kn * NQ + tp, aoff[buf ^ 1], NQ);
      tdm_load_tile16x16(G  + (size_t)kn * NQ + tq, boff[buf ^ 1], NQ);
    }
    const float* As = Ab[buf];
    const float* Bs = Bb[buf];
#pragma unroll
    for (int kk = 0; kk < 16; kk += 4) {
      v2f a, b;
      a.x = As[(kk + 2 * hh)     * 16 + l16];
      a.y = As[(kk + 2 * hh + 1) * 16 + l16];
      b.x = Bs[(kk + 2 * hh)     * 16 + l16];
      b.y = Bs[(kk + 2 * hh + 1) * 16 + l16];
      acc = __builtin_amdgcn_wmma_f32_16x16x4_f32(
          /*neg_a=*/false, a, /*neg_b=*/false, b,
          /*c_mod=*/(short)0, acc, /*reuse_a=*/false, /*reuse_b=*/false);
    }
    buf ^= 1;
  }
#pragma unroll
  for (int r = 0; r < 8; ++r) {
    int row = tp + r + 8 * hh;
    int col = tq + l16;
    KKT[(size_t)row * KLDA + col] = acc[r] + ((row == col) ? 1.0f : 0.0f);
  }
}

// ---------------- 513x513 LU (no pivoting; quasi-definite KKT) + back-sub ------
__global__ void k_solve(float* __restrict__ KKT, float* __restrict__ rhs,
                        float* __restrict__ sol) {
  __shared__ float piv[NK];
  __shared__ float prhs;
  int tid = threadIdx.x, nt = blockDim.x;
  for (int j = 0; j < NK; ++j) {
    __syncthreads();
    for (int c = j + tid; c < NK; c += nt) piv[c] = KKT[(size_t)j * KLDA + c];
    if (tid == 0) prhs = rhs[j];
    __syncthreads();
    float ip = 1.0f / piv[j];
    for (int i = j + 1 + tid; i < NK; i += nt) {
      float f = KKT[(size_t)i * KLDA + j] * ip;
      for (int c = j + 1; c < NK; ++c)
        KKT[(size_t)i * KLDA + c] -= f * piv[c];
      KKT[(size_t)i * KLDA + j] = 0.0f;
      rhs[i] -= f * prhs;
    }
  }
  __syncthreads();
  if (tid == 0) {
    for (int j = NK - 1; j >= 0; --j) {
      float acc = rhs[j];
      for (int c = j + 1; c < NK; ++c) acc -= KKT[(size_t)j * KLDA + c] * sol[c];
      sol[j] = acc / KKT[(size_t)j * KLDA + j];
    }
  }
}

// ---------------- ds = -r_in - G dx ; dz = rv + w*(G dx) -----------------------
__global__ void k_step1(const float* __restrict__ G, const float* __restrict__ sol,
                        const float* __restrict__ r_in, const float* __restrict__ w,
                        const float* __restrict__ rv,
                        float* __restrict__ ds, float* __restrict__ dz) {
  int r = blockIdx.x * blockDim.x + threadIdx.x;
  if (r >= M_TOT) return;
  float acc = 0.0f;
  for (int c = 0; c < NQ; ++c) acc = fmaf(G[(size_t)r * NQ + c], sol[c], acc);
  ds[r] = -r_in[r] - acc;
  dz[r] = rv[r] + w[r] * acc;
}

// ---------------- alpha = min(1, 0.99*min(maxstep(s,ds), maxstep(z,dz))) -------
__global__ void k_step2(const float* __restrict__ sol, const float* __restrict__ ds,
                        const float* __restrict__ dz,
                        float* __restrict__ x, float* __restrict__ y,
                        float* __restrict__ s, float* __restrict__ z) {
  __shared__ float red[1024];
  int tid = threadIdx.x, nt = blockDim.x;
  float am = BIG_C;
  for (int i = tid; i < M_TOT; i += nt) {
    float dsi = ds[i], dzi = dz[i];
    if (dsi < 0.0f) am = fminf(am, -s[i] / dsi);
    if (dzi < 0.0f) am = fminf(am, -z[i] / dzi);
  }
  red[tid] = am;
  __syncthreads();
  for (int off = nt >> 1; off > 0; off >>= 1) {
    if (tid < off) red[tid] = fminf(red[tid], red[tid + off]);
    __syncthreads();
  }
  float alpha = fminf(1.0f, 0.99f * red[0]);
  for (int i = tid; i < M_TOT; i += nt) {
    s[i] += alpha * ds[i];
    z[i] += alpha * dz[i];
  }
  for (int p = tid; p < NQ; p += nt) x[p] += alpha * sol[p];
  if (tid == 0) y[0] += alpha * sol[NQ];
}

__global__ void k_copy_out(const float* __restrict__ x, float* __restrict__ out) {
  int p = blockIdx.x * blockDim.x + threadIdx.x;
  if (p < NQ) out[p] = x[p];
}

extern "C" void kernel_launch(void* const* d_in, const int* in_sizes, int n_in,
                              void* d_out, int out_size, void* d_ws, size_t ws_size,
                              hipStream_t stream) {
  (void)in_sizes; (void)n_in; (void)out_size; (void)ws_size;
  const float* state = (const float*)d_in[0];
  const float* Aeq   = (const float*)d_in[1];
  const float* beq   = (const float*)d_in[2];
  const float* A     = (const float*)d_in[3];
  const float* b     = (const float*)d_in[4];
  const float* ub    = (const float*)d_in[5];
  const float* lb    = (const float*)d_in[6];
  const float* W1    = (const float*)d_in[7];
  const float* b1    = (const float*)d_in[8];
  const float* W2    = (const float*)d_in[9];
  const float* b2    = (const float*)d_in[10];
  const float* W3    = (const float*)d_in[11];
  const float* b3    = (const float*)d_in[12];
  float* out = (float*)d_out;

  float* ws = (float*)d_ws;
  size_t o = 0;
  float* act1 = ws + o; o += HID;          // 4096
  float* act2 = ws + o; o += HID;          // 4096
  float* q    = ws + o; o += NQ;           // 512 (= -raw)
  float* G    = ws + o; o += (size_t)M_TOT * NQ;
  float* Gw   = ws + o; o += (size_t)M_TOT * NQ;
  float* h    = ws + o; o += M_TOT;
  float* x    = ws + o; o += NQ;
  float* y    = ws + o; o += 16;
  float* s    = ws + o; o += M_TOT;
  float* z    = ws + o; o += M_TOT;
  float* rin  = ws + o; o += M_TOT;
  float* req  = ws + o; o += 16;
  float* w    = ws + o; o += M_TOT;
  float* rv   = ws + o; o += M_TOT;
  float* rhs  = ws + o; o += 544;
  float* KKT  = ws + o; o += (size_t)KLDA * KLDA;
  float* sol  = ws + o; o += 544;
  float* ds   = ws + o; o += M_TOT;
  float* dz   = ws + o; o += M_TOT;

  // ---- MLP: relu(state@W1+b1) -> relu(@W2+b2) -> q = -(@W3+b3) ----
  k_gemv<<<(HID + 255) / 256, 256, 0, stream>>>(state, W1, b1, act1, DIMS, HID, 1);
  k_gemv<<<(HID + 255) / 256, 256, 0, stream>>>(act1, W2, b2, act2, HID, HID, 1);
  k_gemv<<<(NQ + 255) / 256, 256, 0, stream>>>(act2, W3, b3, q, HID, NQ, 2);

  // ---- QP setup ----
  int gebuild = (M_TOT * NQ + 255) / 256;
  k_build<<<gebuild, 256, 0, stream>>>(A, b, ub, lb, G, h, x, y, s, z);

  // ---- IPM iterations ----
  int grows = (M_TOT + 255) / 256;
  for (int it = 0; it < QP_ITERS; ++it) {
    k_rin<<<grows, 256, 0, stream>>>(G, x, s, h, Aeq, beq, rin, req);
    k_muwrv<<<1, 1024, 0, stream>>>(s, z, rin, w, rv);
    k_rhs_border<<<1, 544, 0, stream>>>(G, z, rv, x, q, Aeq, y, req, rhs, KKT);
    k_scale_g<<<gebuild, 256, 0, stream>>>(G, w, Gw);
    k_gtwg_wmma<<<(NQ / 16) * (NQ / 16) / 8, 256, 0, stream>>>(Gw, G, KKT);
    k_solve<<<1, 1024, 0, stream>>>(KKT, rhs, sol);
    k_step1<<<grows, 256, 0, stream>>>(G, sol, rin, w, rv, ds, dz);
    k_step2<<<1, 1024, 0, stream>>>(sol, ds, dz, x, y, s, z);
  }

  k_copy_out<<<(NQ + 255) / 256, 256, 0, stream>>>(x, out);
}